// LReLu_torch_72215580115320
// MI455X (gfx1250) — compile-verified
//
#include <hip/hip_runtime.h>

// ---------------------------------------------------------------------------
// Fused: 2x bilinear upsample -> LeakyReLU -> (256->128->64 antialias resize
// composed into one 64x256 matrix R, applied on rows and cols via WMMA f32)
// -> + per-channel bias.   x:(8,128,128,128) f32  ->  out:(8,128,64,64) f32
// ---------------------------------------------------------------------------

typedef __attribute__((ext_vector_type(2))) float v2f;
typedef __attribute__((ext_vector_type(8))) float v8f;

#define SRR 260   // LDS row stride of R   (260 % 64 == 4 -> conflict-free lane gathers)
#define SUS 48    // LDS row stride of U block (48*2 % 64 == 32 -> halves disjoint)
#define STS 36    // LDS row stride of T block (36*l % 64 = distinct multiples of 4 -> conflict-free)

#define XBUF_FLOATS (128 * 32)

#define OFF_R   0
#define OFF_X0  (OFF_R + 64 * SRR)        // 16640 : X slice buffers (2 x 128x32)
#define OFF_U   (OFF_X0 + 2 * XBUF_FLOATS) // 24832 : U block (256 x SUS)
#define OFF_T   (OFF_U + 256 * SUS)       // 37120 : T block (64 x STS)
#define LDS_FLOATS (OFF_T + 64 * STS)     // 39424 floats = 157696 bytes

// ---- gfx1250 async global->LDS path (guarded; falls back to float4 copy) ----
#if __has_builtin(__builtin_amdgcn_global_load_async_to_lds_b128)
#define HAVE_ASYNC_LDS 1
typedef __attribute__((__vector_size__(16))) int v4i;
typedef __attribute__((address_space(1))) v4i gas_v4i;   // global
typedef __attribute__((address_space(3))) v4i las_v4i;   // LDS
#endif

#if __has_builtin(__builtin_amdgcn_s_wait_asynccnt)
#define WAIT_ASYNC() __builtin_amdgcn_s_wait_asynccnt(0)
#else
#define WAIT_ASYNC() asm volatile("s_wait_asynccnt 0" ::: "memory")
#endif

__device__ __forceinline__ v8f wmma_f32_16x16x4(v2f a, v2f b, v8f c) {
  // D = A(16x4,f32) * B(4x16,f32) + C(16x16,f32)
  return __builtin_amdgcn_wmma_f32_16x16x4_f32(
      /*neg_a=*/false, a, /*neg_b=*/false, b,
      /*c_mod=*/(short)0, c, /*reuse_a=*/false, /*reuse_b=*/false);
}

// Stage one 128x32 column slice of X into LDS buffer. Addresses are clamped at
// 4-column-group granularity; clamped groups only land in slice columns that
// the U-fill provably never reads (used range is qi in [7,24]).
__device__ __forceinline__ void stage_x_slice(const float* __restrict__ xg,
                                              float* __restrict__ sXdst,
                                              int qbase, int tid) {
#pragma unroll
  for (int k = 0; k < 4; ++k) {
    int e = tid + 256 * k;            // 0..1023
    int p = e >> 3;                   // row 0..127
    int g = e & 7;                    // 4-column group 0..7
    int qg = qbase + 4 * g;
    qg = qg < 0 ? 0 : (qg > 124 ? 124 : qg);
    const float* src = xg + (size_t)p * 128 + qg;
    float* dst = sXdst + p * 32 + 4 * g;
#if defined(HAVE_ASYNC_LDS)
    __builtin_amdgcn_global_load_async_to_lds_b128(
        (gas_v4i*)src, (las_v4i*)dst, /*offset=*/0, /*cpol=*/0);
#else
    *(float4*)dst = *(const float4*)src;
#endif
  }
}

// Build R = M2(64x128) @ M1(128x256): composition of the two antialiased
// triangle (PIL bilinear) downsample matrices, with per-row normalization.
__global__ void __launch_bounds__(256) build_R_kernel(float* __restrict__ Rm) {
  int e = blockIdx.x * 256 + threadIdx.x;   // 0..16383
  int o = e >> 8;                           // output row   0..63
  int i = e & 255;                          // input column 0..255
  float c2 = 2.0f * (float)o + 0.5f;        // center of M2 row o over 128
  float s2 = 0.0f;
  for (int d = -2; d <= 3; ++d) {
    int k = 2 * o + d;
    if (k >= 0 && k < 128)
      s2 += fmaxf(0.0f, 1.0f - fabsf((float)k - c2) * 0.5f);
  }
  float acc = 0.0f;
  for (int d = -2; d <= 3; ++d) {
    int k = 2 * o + d;
    if (k < 0 || k >= 128) continue;
    float w2 = fmaxf(0.0f, 1.0f - fabsf((float)k - c2) * 0.5f);
    if (w2 <= 0.0f) continue;
    float c1 = 2.0f * (float)k + 0.5f;      // center of M1 row k over 256
    float s1 = 0.0f;
    for (int dd = -2; dd <= 3; ++dd) {
      int ii = 2 * k + dd;
      if (ii >= 0 && ii < 256)
        s1 += fmaxf(0.0f, 1.0f - fabsf((float)ii - c1) * 0.5f);
    }
    float w1 = fmaxf(0.0f, 1.0f - fabsf((float)i - c1) * 0.5f);
    acc += (w2 / s2) * (w1 / s1);
  }
  Rm[e] = acc;
}

__global__ void __launch_bounds__(256) fused_resize_lrelu_kernel(
    const float* __restrict__ x, const float* __restrict__ bias,
    const float* __restrict__ Rm, float* __restrict__ out) {
  extern __shared__ float smem[];
  float* sR = smem + OFF_R;       // 64 x 256 (stride SRR)
  float* sU = smem + OFF_U;       // 256 x 32 LeakyReLU(upsample) block
  float* sT = smem + OFF_T;       // 64 x 32  T = R*U block

  const int tid = threadIdx.x;
  const int img = blockIdx.x;                       // b*128 + c
  const float* xg = x + (size_t)img * (128 * 128);

  // ---- stage R into LDS (padded rows) ----
#pragma unroll
  for (int k = 0; k < 64; ++k) {
    int e = tid + 256 * k;
    sR[(e >> 8) * SRR + (e & 255)] = Rm[e];
  }
  // ---- prologue: stage slice for block 0 into buffer 0 ----
  stage_x_slice(xg, smem + OFF_X0, /*qbase=*/-8, tid);
  WAIT_ASYNC();

  const int wave = tid >> 5;
  const int lane = tid & 31;
  const int half = lane >> 4;       // selects K pair / M+8 rows
  const int l    = lane & 15;

  const int m0  = (wave & 3) * 16;  // M tile (shared by GEMM1 & GEMM2)
  const int nt1 = (wave >> 2);      // GEMM1 N tile: 0..1
  const int n0a = (wave >> 2) * 16; // GEMM2 N tiles: {0,16} and {32,48}
  const int n0b = n0a + 32;

  // per-thread upsample-column constants for the U fill
  const int jj = tid & 31;          // U column within block
  const int i0 = (tid >> 5) * 32;   // U row range start

  v8f accY0 = {};                   // Y tile accumulators (live across blocks)
  v8f accY1 = {};

  __syncthreads();

  for (int jb = 0; jb < 8; ++jb) {
    const int qbase = jb * 16 - 8;  // X column covered by slice index 0
    const float* sX = smem + OFF_X0 + (jb & 1) * XBUF_FLOATS;

    // ---- fill U block: LeakyReLU(2x bilinear upsample), weights 0.25/0.75 ----
    {
      const int j = jb * 32 + jj;
      int q0 = (j - 1) >> 1;        // floor((j-1)/2)
      int q1 = q0 + 1;
      const float fw = (j & 1) ? 0.25f : 0.75f;
      q0 = q0 < 0 ? 0 : q0;
      q1 = q1 > 127 ? 127 : q1;
      const int qi0 = q0 - qbase, qi1 = q1 - qbase;
      const float ww0 = 1.0f - fw, ww1 = fw;
#pragma unroll 4
      for (int it = 0; it < 32; ++it) {
        int i = i0 + it;
        int p0 = (i - 1) >> 1;
        int p1 = p0 + 1;
        float fh = (i & 1) ? 0.25f : 0.75f;
        p0 = p0 < 0 ? 0 : p0;
        p1 = p1 > 127 ? 127 : p1;
        const float* r0 = sX + p0 * 32;
        const float* r1 = sX + p1 * 32;
        float top = r0[qi0] * ww0 + r0[qi1] * ww1;
        float bot = r1[qi0] * ww0 + r1[qi1] * ww1;
        float u = top * (1.0f - fh) + bot * fh;
        u = (u >= 0.0f) ? u : 0.01f * u;            // LeakyReLU(0.01)
        sU[i * SUS + jj] = u;
      }
    }
    __syncthreads();

    // ---- software pipeline: kick off next block's X slice (async -> LDS),
    //      latency hides under GEMM1 + GEMM2 below ----
    if (jb + 1 < 8)
      stage_x_slice(xg, smem + OFF_X0 + ((jb + 1) & 1) * XBUF_FLOATS,
                    qbase + 16, tid);

    // ---- GEMM1: T_blk(64x32) = R(64x256) @ U_blk(256x32), K=256,
    //      split into two independent K=128 chains for WMMA ILP ----
    {
      v8f accA = {};
      v8f accB = {};
      const float* aBase = sR + (m0 + l) * SRR + 2 * half;
      const float* bBase = sU + (2 * half) * SUS + nt1 * 16 + l;
#pragma unroll 4
      for (int k0 = 0; k0 < 128; k0 += 4) {
        v2f a0 = *(const v2f*)(aBase + k0);
        v2f a1 = *(const v2f*)(aBase + 128 + k0);
        v2f b0, b1;
        b0.x = bBase[k0 * SUS];
        b0.y = bBase[(k0 + 1) * SUS];
        b1.x = bBase[(k0 + 128) * SUS];
        b1.y = bBase[(k0 + 129) * SUS];
        accA = wmma_f32_16x16x4(a0, b0, accA);
        accB = wmma_f32_16x16x4(a1, b1, accB);
      }
#pragma unroll
      for (int v = 0; v < 8; ++v)
        sT[(m0 + v + 8 * half) * STS + nt1 * 16 + l] = accA[v] + accB[v];
    }
    __syncthreads();

    // ---- GEMM2 partial: Y(64x64) += T_blk(64x32) @ R^T_blk(32x64), K=32 ----
    {
      const float* aBase  = sT + (m0 + l) * STS + 2 * half;
      const float* b0Base = sR + (n0a + l) * SRR + jb * 32 + 2 * half;
      const float* b1Base = sR + (n0b + l) * SRR + jb * 32 + 2 * half;
#pragma unroll
      for (int k0 = 0; k0 < 32; k0 += 4) {
        v2f a  = *(const v2f*)(aBase + k0);
        v2f b0 = *(const v2f*)(b0Base + k0);
        v2f b1 = *(const v2f*)(b1Base + k0);
        accY0 = wmma_f32_16x16x4(a, b0, accY0);
        accY1 = wmma_f32_16x16x4(a, b1, accY1);
      }
    }
    // own async loads into the other slice buffer must be in LDS before the
    // barrier releases the next fill
    WAIT_ASYNC();
    __syncthreads();
  }

  // ---- epilogue: + bias, store ----
  {
    const float bv = bias[img & 127];
    float* obase = out + (size_t)img * 4096;
#pragma unroll
    for (int v = 0; v < 8; ++v) {
      int row = m0 + v + 8 * half;
      obase[row * 64 + n0a + l] = accY0[v] + bv;
      obase[row * 64 + n0b + l] = accY1[v] + bv;
    }
  }
}

extern "C" void kernel_launch(void* const* d_in, const int* in_sizes, int n_in,
                              void* d_out, int out_size, void* d_ws, size_t ws_size,
                              hipStream_t stream) {
  (void)in_sizes; (void)n_in; (void)out_size; (void)ws_size;
  const float* x    = (const float*)d_in[0];
  const float* bias = (const float*)d_in[1];
  float* Rm  = (float*)d_ws;           // 64*256 f32 = 64 KB scratch
  float* out = (float*)d_out;

  // Rebuilt every call: deterministic, immune to workspace poisoning.
  build_R_kernel<<<64, 256, 0, stream>>>(Rm);

  const size_t lds_bytes = (size_t)LDS_FLOATS * sizeof(float);  // 157696 B
  (void)hipFuncSetAttribute((const void*)fused_resize_lrelu_kernel,
                            hipFuncAttributeMaxDynamicSharedMemorySize,
                            (int)lds_bytes);
  fused_resize_lrelu_kernel<<<8 * 128, 256, lds_bytes, stream>>>(x, bias, Rm, out);
}